// TrueBiomimeticGNN_50319836840367
// MI455X (gfx1250) — compile-verified
//
#include <hip/hip_runtime.h>

typedef __attribute__((ext_vector_type(2))) float v2f;
typedef __attribute__((ext_vector_type(8))) float v8f;

#define N_NODES 10000
#define N_EDGES 320000
#define BATCH   16
#define FEAT    128
#define C1      16
#define C2      32
#define NCLS    10

// ---------------- normalization ----------------

__global__ void k_init_deg(float* deg) {
  int n = blockIdx.x * blockDim.x + threadIdx.x;
  if (n < N_NODES) deg[n] = 1.0f;   // self-loop weight 1
}

__global__ void k_scatter_deg(const int* ei, const float* ew, float* deg) {
  int e = blockIdx.x * blockDim.x + threadIdx.x;
  if (e < N_EDGES) atomicAdd(&deg[ei[N_EDGES + e]], ew[e]);
}

__global__ void k_dis_selfc(float* dis /*in: deg*/, float* selfc) {
  int n = blockIdx.x * blockDim.x + threadIdx.x;
  if (n < N_NODES) {
    float d = dis[n];
    dis[n]   = rsqrtf(d);   // in-place deg -> dis
    selfc[n] = 1.0f / d;    // dis*dis
  }
}

__global__ void k_norm(const int* ei, const float* ew, const float* dis, float* norm) {
  int e = blockIdx.x * blockDim.x + threadIdx.x;
  if (e < N_EDGES)
    norm[e] = dis[ei[e]] * ew[e] * dis[ei[N_EDGES + e]];
}

// ---------------- encoder: h0 = x @ enc_W + enc_b  (WMMA f32 16x16x4) ----------------
// One wave per 16-node tile. A = x[16,128], B = enc_W[:,tile], D = [batch=16][node=16].
// Also seeds agg1 with the self-loop term selfc[n]*h0.

__global__ void __launch_bounds__(32) k_encoder(const float* __restrict__ x,
                                                const float* __restrict__ W,
                                                const float* __restrict__ bias,
                                                const float* __restrict__ selfc,
                                                float* __restrict__ h0,
                                                float* __restrict__ agg1) {
  int lane = threadIdx.x;
  int m    = lane & 15;          // A row / D col-group
  int kh   = (lane >> 4) * 2;    // K half select (ISA §7.12.2)
  int nn   = blockIdx.x * 16 + m;

  v8f acc = {};
  for (int k0 = 0; k0 < FEAT; k0 += 4) {
    v2f a, b;
    a.x = x[m * FEAT + k0 + kh];
    a.y = x[m * FEAT + k0 + kh + 1];
    b.x = W[(k0 + kh) * N_NODES + nn];
    b.y = W[(k0 + kh + 1) * N_NODES + nn];
    acc = __builtin_amdgcn_wmma_f32_16x16x4_f32(false, a, false, b,
                                                (short)0, acc, false, false);
  }
  // lane holds D[M = j + 8*(lane>>4)][N = lane&15] in acc[j]
  float bb   = bias[nn];
  float sc   = selfc[nn];
  int   base = nn * BATCH + (lane >> 4) * 8;   // h0 layout [node][batch]
#pragma unroll
  for (int j = 0; j < 8; ++j) {
    float v = acc[j] + bb;
    h0[base + j]   = v;
    agg1[base + j] = sc * v;    // self-loop seed for layer-1 aggregation
  }
}

// ---------------- layer 1: rank-1 message aggregation on scalars ----------------

__global__ void k_edge1(const int* __restrict__ ei, const float* __restrict__ norm,
                        const float* __restrict__ h0, float* __restrict__ agg1) {
  int t = blockIdx.x * blockDim.x + threadIdx.x;  // (edge, batch)
  if (t >= N_EDGES * BATCH) return;
  int e = t >> 4;
  int b = t & 15;
  int r = ei[e];
  int c = ei[N_EDGES + e];
  atomicAdd(&agg1[c * BATCH + b], norm[e] * h0[r * BATCH + b]);
}

// pointwise: h1 = relu(agg1 * W1 + b1); also seeds agg2 with selfc*h1
__global__ void k_layer1(const float* __restrict__ agg1, const float* __restrict__ selfc,
                         const float* __restrict__ W1, const float* __restrict__ b1,
                         float* __restrict__ h1, float* __restrict__ agg2) {
  int t = blockIdx.x * blockDim.x + threadIdx.x;  // (node, batch)
  if (t >= N_NODES * BATCH) return;
  int   n  = t >> 4;
  float v  = agg1[t];
  float sc = selfc[n];
#pragma unroll
  for (int c = 0; c < C1; ++c) {
    float h = fmaf(v, W1[c], b1[c]);
    h = h > 0.0f ? h : 0.0f;
    h1[t * C1 + c]   = h;
    agg2[t * C1 + c] = sc * h;
  }
}

// ---------------- layer 2: edge aggregation of 16 channels ----------------

__global__ void __launch_bounds__(256) k_edge2(const int* __restrict__ ei,
                                               const float* __restrict__ norm,
                                               const float* __restrict__ h1,
                                               float* __restrict__ agg2) {
  int i  = threadIdx.x;                 // (batch,channel) 0..255
  int e0 = blockIdx.x * 8;
#pragma unroll
  for (int q = 0; q < 8; ++q) {
    int   e  = e0 + q;
    int   r  = ei[e];
    int   c  = ei[N_EDGES + e];
    float nv = norm[e];
    atomicAdd(&agg2[c * (BATCH * C1) + i], nv * h1[r * (BATCH * C1) + i]);
  }
}

__global__ void k_zero_pooled(float* pooled) {
  int t = blockIdx.x * blockDim.x + threadIdx.x;
  if (t < BATCH * C2) pooled[t] = 0.0f;
}

// ---------------- layer-2 GEMM + bias + ReLU + mean-pool (WMMA f32 16x16x4) ----------------
// Per node n: D[16x32] = agg2[n] (16 batches x 16 ch) @ W2[16,32]; fused into pooled.

__global__ void __launch_bounds__(32) k_gemm2_pool(const float* __restrict__ agg2,
                                                   const float* __restrict__ W2,
                                                   const float* __restrict__ b2,
                                                   float* __restrict__ pooled) {
  int lane = threadIdx.x;
  int m    = lane & 15;
  int kh   = (lane >> 4) * 2;

  // B fragments: invariant across the node loop, hoisted.
  v2f bf[2][4];
#pragma unroll
  for (int t = 0; t < 2; ++t)
#pragma unroll
    for (int ks = 0; ks < 4; ++ks) {
      int c  = t * 16 + m;
      int k0 = ks * 4;
      bf[t][ks].x = W2[(k0 + kh) * C2 + c];
      bf[t][ks].y = W2[(k0 + kh + 1) * C2 + c];
    }
  float bias0 = b2[m];
  float bias1 = b2[16 + m];

  v8f pool0 = {}, pool1 = {};
  for (int n = blockIdx.x; n < N_NODES; n += gridDim.x) {
    const float* Arow = agg2 + n * (BATCH * C1);
    v8f acc0 = {}, acc1 = {};
#pragma unroll
    for (int ks = 0; ks < 4; ++ks) {
      int k0 = ks * 4;
      v2f a;
      a.x = Arow[m * C1 + k0 + kh];
      a.y = Arow[m * C1 + k0 + kh + 1];
      acc0 = __builtin_amdgcn_wmma_f32_16x16x4_f32(false, a, false, bf[0][ks],
                                                   (short)0, acc0, false, false);
      acc1 = __builtin_amdgcn_wmma_f32_16x16x4_f32(false, a, false, bf[1][ks],
                                                   (short)0, acc1, false, false);
    }
#pragma unroll
    for (int j = 0; j < 8; ++j) {
      float v0 = acc0[j] + bias0; pool0[j] += (v0 > 0.0f ? v0 : 0.0f);
      float v1 = acc1[j] + bias1; pool1[j] += (v1 > 0.0f ? v1 : 0.0f);
    }
  }
  int b = (lane >> 4) * 8;   // batch row base
#pragma unroll
  for (int j = 0; j < 8; ++j) {
    atomicAdd(&pooled[(b + j) * C2 + m],       pool0[j]);
    atomicAdd(&pooled[(b + j) * C2 + 16 + m],  pool1[j]);
  }
}

// ---------------- final MLP: [16,32] -> relu[16,16] -> [16,10] ----------------

__global__ void __launch_bounds__(256) k_final(const float* __restrict__ pooled,
                                               const float* __restrict__ c1W,
                                               const float* __restrict__ c1b,
                                               const float* __restrict__ c2W,
                                               const float* __restrict__ c2b,
                                               float* __restrict__ out) {
  __shared__ float ps[BATCH * C2];
  __shared__ float zs[BATCH * 16];
  int tid = threadIdx.x;
  for (int i = tid; i < BATCH * C2; i += 256) ps[i] = pooled[i] * (1.0f / N_NODES);
  __syncthreads();
  {
    int b = tid >> 4, j = tid & 15;
    float s = c1b[j];
    for (int c = 0; c < C2; ++c) s = fmaf(ps[b * C2 + c], c1W[c * 16 + j], s);
    zs[tid] = s > 0.0f ? s : 0.0f;
  }
  __syncthreads();
  if (tid < BATCH * NCLS) {
    int b = tid / NCLS, k = tid % NCLS;
    float s = c2b[k];
    for (int j = 0; j < 16; ++j) s = fmaf(zs[b * 16 + j], c2W[j * NCLS + k], s);
    out[tid] = s;
  }
}

// ---------------- launcher ----------------

extern "C" void kernel_launch(void* const* d_in, const int* in_sizes, int n_in,
                              void* d_out, int out_size, void* d_ws, size_t ws_size,
                              hipStream_t stream) {
  (void)in_sizes; (void)n_in; (void)out_size; (void)ws_size;

  const float* x     = (const float*)d_in[0];
  const int*   ei    = (const int*)  d_in[1];
  const float* ew    = (const float*)d_in[2];
  const float* enc_W = (const float*)d_in[3];
  const float* enc_b = (const float*)d_in[4];
  const float* W1    = (const float*)d_in[5];
  const float* b1    = (const float*)d_in[6];
  const float* W2    = (const float*)d_in[7];
  const float* b2    = (const float*)d_in[8];
  const float* c1W   = (const float*)d_in[9];
  const float* c1b   = (const float*)d_in[10];
  const float* c2W   = (const float*)d_in[11];
  const float* c2b   = (const float*)d_in[12];
  float* out = (float*)d_out;

  float* ws = (float*)d_ws;
  size_t off = 0;
  float* dis    = ws + off; off += N_NODES;            // deg -> dis in place
  float* selfc  = ws + off; off += N_NODES;
  float* norm   = ws + off; off += N_EDGES;
  float* h0     = ws + off; off += (size_t)N_NODES * BATCH;
  float* agg1   = ws + off; off += (size_t)N_NODES * BATCH;
  float* h1     = ws + off; off += (size_t)N_NODES * BATCH * C1;
  float* agg2   = ws + off; off += (size_t)N_NODES * BATCH * C1;
  float* pooled = ws + off; off += BATCH * C2;

  k_init_deg   <<<(N_NODES + 255) / 256, 256, 0, stream>>>(dis);
  k_scatter_deg<<<(N_EDGES + 255) / 256, 256, 0, stream>>>(ei, ew, dis);
  k_dis_selfc  <<<(N_NODES + 255) / 256, 256, 0, stream>>>(dis, selfc);
  k_norm       <<<(N_EDGES + 255) / 256, 256, 0, stream>>>(ei, ew, dis, norm);

  k_encoder<<<N_NODES / 16, 32, 0, stream>>>(x, enc_W, enc_b, selfc, h0, agg1);

  k_edge1 <<<(N_EDGES * BATCH) / 256, 256, 0, stream>>>(ei, norm, h0, agg1);
  k_layer1<<<(N_NODES * BATCH + 255) / 256, 256, 0, stream>>>(agg1, selfc, W1, b1, h1, agg2);
  k_edge2 <<<N_EDGES / 8, 256, 0, stream>>>(ei, norm, h1, agg2);

  k_zero_pooled<<<2, 256, 0, stream>>>(pooled);
  k_gemm2_pool <<<2000, 32, 0, stream>>>(agg2, W2, b2, pooled);
  k_final      <<<1, 256, 0, stream>>>(pooled, c1W, c1b, c2W, c2b, out);
}